// GATModel_10788957848006
// MI455X (gfx1250) — compile-verified
//
#include <hip/hip_runtime.h>
#include <cstdint>
#include <cstddef>

typedef __attribute__((ext_vector_type(2))) float v2f;
typedef __attribute__((ext_vector_type(8))) float v8f;

static constexpr int NNODES = 50000;
static constexpr int HEADS  = 4;
static constexpr int C1     = 128;
static constexpr int C2     = 64;
static constexpr float NEG_SLOPE = 0.2f;

// ---------------------------------------------------------------------------
// float atomic max via ordered-int trick (lowers to global_atomic_max/min)
// ---------------------------------------------------------------------------
__device__ __forceinline__ void atomicMaxF(float* addr, float val) {
    if (val >= 0.f) atomicMax((int*)addr, __float_as_int(val));
    else            atomicMin((unsigned int*)addr, (unsigned int)__float_as_int(val));
}

__global__ void fill_f32(float* __restrict__ p, float v, size_t n) {
    size_t i = (size_t)blockIdx.x * blockDim.x + threadIdx.x;
    if (i < n) p[i] = v;
}

// ---------------------------------------------------------------------------
// GEMM: C[M, Nreal] = A[M, K] @ B[K, Nreal] (+ optional bias row)
// FP32 WMMA 16x16x4; each wave computes a 16 x (16*NSUB) strip: one shared A
// fragment feeds NSUB accumulators.
//
// Inner loop: carries explicit row pointers (no per-iteration index*stride
// multiply), gathers ALL B fragments before issuing the WMMA chain so every
// load is in flight before the first s_wait.
//
// GUARD=false: Nreal == Npad; B columns addressed as immediate offsets.
// GUARD=true : per-subtile clamped pointers + mask multiply (no EXEC games).
//
// VGPR layouts (ISA 7.12.2, wave32):
//   A 16x4 f32 : lane&15 = M row; lanes 0-15 -> {K=0,1}, lanes 16-31 -> {K=2,3}
//   B 4x16 f32 : VGPR j : K = 2*half + j, N = lane&15
//   C/D 16x16  : VGPR r : M = r + 8*half,  N = lane&15
// ---------------------------------------------------------------------------
template<int NSUB, bool GUARD>
__global__ void gemm16_wmma_f32(const float* __restrict__ A,
                                const float* __restrict__ B,
                                const float* __restrict__ bias,
                                float* __restrict__ C,
                                int M, int K, int Nreal, int Npad) {
    const int lane    = threadIdx.x & 31;
    const int wave    = threadIdx.x >> 5;
    const int nGroups = Npad / (16 * NSUB);
    const int mTiles  = M >> 4;
    int tile = blockIdx.x * (blockDim.x >> 5) + wave;
    if (tile >= mTiles * nGroups) return;         // wave-uniform
    const int mt = tile / nGroups;
    const int ng = tile - mt * nGroups;
    const int m0 = mt << 4;
    const int n0 = ng * (16 * NSUB);
    const int ln   = lane & 15;
    const int half = lane >> 4;

    v8f acc[NSUB];
    #pragma unroll
    for (int j = 0; j < NSUB; ++j) acc[j] = (v8f){};

    // A fragment pointer: row m0+ln, starting at K-index 2*half, step 4
    const float* aPtr = A + (size_t)(m0 + ln) * K + 2 * half;
    const long long rowStep = 4LL * Nreal;        // k-step advance in floats

    if (!GUARD) {
        // ------- branch-free path: immediate-offset B addressing -------
        const float* p0 = B + (size_t)(2 * half) * Nreal + (n0 + ln); // row kb
        const float* p1 = p0 + Nreal;                                 // row kb+1
        for (int k = 0; k < K; k += 4) {
            __builtin_prefetch(aPtr + 32, 0, 1);       // global_prefetch_b8
            const v2f a = *(const v2f*)aPtr;
            v2f bf[NSUB];
            #pragma unroll
            for (int j = 0; j < NSUB; ++j) {           // all loads issued first
                bf[j].x = p0[16 * j];
                bf[j].y = p1[16 * j];
            }
            #pragma unroll
            for (int j = 0; j < NSUB; ++j)             // then the WMMA chain
                acc[j] = __builtin_amdgcn_wmma_f32_16x16x4_f32(
                             false, a, false, bf[j], (short)0, acc[j], false, false);
            aPtr += 4;
            p0 += rowStep;
            p1 += rowStep;
        }
    } else {
        // ------- guarded path (classifier only): clamped cols + mask -------
        const float* q0[NSUB];
        const float* q1[NSUB];
        float mk[NSUB];
        #pragma unroll
        for (int j = 0; j < NSUB; ++j) {
            const int col  = n0 + 16 * j + ln;
            const int colS = col < Nreal ? col : Nreal - 1;
            mk[j] = col < Nreal ? 1.f : 0.f;
            q0[j] = B + (size_t)(2 * half) * Nreal + colS;
            q1[j] = q0[j] + Nreal;
        }
        for (int k = 0; k < K; k += 4) {
            const v2f a = *(const v2f*)aPtr;
            v2f bf[NSUB];
            #pragma unroll
            for (int j = 0; j < NSUB; ++j) {
                bf[j].x = q0[j][0] * mk[j];
                bf[j].y = q1[j][0] * mk[j];
            }
            #pragma unroll
            for (int j = 0; j < NSUB; ++j)
                acc[j] = __builtin_amdgcn_wmma_f32_16x16x4_f32(
                             false, a, false, bf[j], (short)0, acc[j], false, false);
            aPtr += 4;
            #pragma unroll
            for (int j = 0; j < NSUB; ++j) { q0[j] += rowStep; q1[j] += rowStep; }
        }
    }

    #pragma unroll
    for (int j = 0; j < NSUB; ++j) {
        const int col = n0 + 16 * j + ln;
        if (!GUARD || col < Nreal) {
            const float bb = bias ? bias[col] : 0.f;
            #pragma unroll
            for (int r = 0; r < 8; ++r) {
                const int row = m0 + r + 8 * half;
                C[(size_t)row * Nreal + col] = acc[j][r] + bb;
            }
        }
    }
}

// ---------------------------------------------------------------------------
// Per-node attention halves: asrc[n,h] = <h[n,h,:], a_src[h,:]>, same for adst.
// One wave32 per (node, head); shfl_xor tree reduction.
// ---------------------------------------------------------------------------
__global__ void attn_scores(const float* __restrict__ h,
                            const float* __restrict__ a_src,
                            const float* __restrict__ a_dst,
                            float* __restrict__ asrc,
                            float* __restrict__ adst,
                            int nNodes, int C) {
    const int gw   = (int)((blockIdx.x * (size_t)blockDim.x + threadIdx.x) >> 5);
    const int lane = threadIdx.x & 31;
    if (gw >= nNodes * HEADS) return;             // wave-uniform
    const int node = gw >> 2;                     // HEADS == 4
    const int hd   = gw & 3;
    const float* hp = h + (size_t)node * (HEADS * C) + hd * C;
    const float* as = a_src + hd * C;
    const float* ad = a_dst + hd * C;
    float s = 0.f, d = 0.f;
    for (int c = lane; c < C; c += 32) {
        const float v = hp[c];
        s += v * as[c];
        d += v * ad[c];
    }
    #pragma unroll
    for (int off = 16; off; off >>= 1) {
        s += __shfl_xor(s, off, 32);
        d += __shfl_xor(d, off, 32);
    }
    if (lane == 0) { asrc[gw] = s; adst[gw] = d; }
}

// ---------------------------------------------------------------------------
// Edge pass 1: e = leaky_relu(asrc[src] + adst[dst]); segment-max into mmax.
// ---------------------------------------------------------------------------
__global__ void edge_score_max(const int* __restrict__ ei,
                               const float* __restrict__ asrc,
                               const float* __restrict__ adst,
                               float* __restrict__ ebuf,
                               float* __restrict__ mmax,
                               int Eorig, int EE) {
    const int idx = blockIdx.x * blockDim.x + threadIdx.x;
    if (idx >= EE * HEADS) return;
    const int ee = idx >> 2, hd = idx & 3;
    int s, d;
    if (ee < Eorig) { s = ei[ee]; d = ei[Eorig + ee]; }
    else            { s = d = ee - Eorig; }
    float e = asrc[s * HEADS + hd] + adst[d * HEADS + hd];
    e = (e > 0.f) ? e : NEG_SLOPE * e;
    ebuf[idx] = e;
    atomicMaxF(&mmax[d * HEADS + hd], e);
}

// Edge pass 2: ex = exp(e - m[dst]); segment-sum into denom; store ex in place.
__global__ void edge_exp_sum(const int* __restrict__ ei,
                             const float* __restrict__ mmax,
                             float* __restrict__ ebuf,
                             float* __restrict__ denom,
                             int Eorig, int EE) {
    const int idx = blockIdx.x * blockDim.x + threadIdx.x;
    if (idx >= EE * HEADS) return;
    const int ee = idx >> 2, hd = idx & 3;
    int d;
    if (ee < Eorig) d = ei[Eorig + ee];
    else            d = ee - Eorig;
    const float ex = expf(ebuf[idx] - mmax[d * HEADS + hd]);
    ebuf[idx] = ex;
    atomicAdd(&denom[d * HEADS + hd], ex);
}

// Edge pass 3: agg[dst] += h[src] * alpha. One wave32 per (edge, head).
__global__ void edge_aggregate(const int* __restrict__ ei,
                               const float* __restrict__ h,
                               const float* __restrict__ ebuf,
                               const float* __restrict__ denom,
                               float* __restrict__ agg,
                               int Eorig, int EE, int C) {
    const int gw   = (int)(((size_t)blockIdx.x * blockDim.x + threadIdx.x) >> 5);
    const int lane = threadIdx.x & 31;
    if (gw >= EE * HEADS) return;                 // wave-uniform
    const int ee = gw >> 2, hd = gw & 3;
    int s, d;
    if (ee < Eorig) { s = ei[ee]; d = ei[Eorig + ee]; }
    else            { s = d = ee - Eorig; }
    const float alpha = ebuf[gw] / (denom[d * HEADS + hd] + 1e-16f);
    const float* hp = h   + (size_t)s * (HEADS * C) + hd * C;
    float*       op = agg + (size_t)d * (HEADS * C) + hd * C;
    for (int c = lane; c < C; c += 32)
        atomicAdd(op + c, hp[c] * alpha);
}

// y = elu(y + bias[col]), in place over [rows, D]
__global__ void bias_elu(float* __restrict__ y, const float* __restrict__ b,
                         size_t n, int D) {
    const size_t i = (size_t)blockIdx.x * blockDim.x + threadIdx.x;
    if (i >= n) return;
    const int col = (int)(i % (size_t)D);
    const float v = y[i] + b[col];
    y[i] = (v > 0.f) ? v : (expf(v) - 1.f);
}

// ---------------------------------------------------------------------------
// Host orchestration
// ---------------------------------------------------------------------------
static inline int cdiv(long long a, long long b) { return (int)((a + b - 1) / b); }

extern "C" void kernel_launch(void* const* d_in, const int* in_sizes, int n_in,
                              void* d_out, int out_size, void* d_ws, size_t ws_size,
                              hipStream_t stream) {
    const float* x      = (const float*)d_in[0];
    const int*   ei     = (const int*)  d_in[1];
    const float* W1     = (const float*)d_in[2];
    const float* a_src1 = (const float*)d_in[3];
    const float* a_dst1 = (const float*)d_in[4];
    const float* b1     = (const float*)d_in[5];
    const float* W2     = (const float*)d_in[6];
    const float* a_src2 = (const float*)d_in[7];
    const float* a_dst2 = (const float*)d_in[8];
    const float* b2     = (const float*)d_in[9];
    const float* Wf     = (const float*)d_in[10];
    const float* bf     = (const float*)d_in[11];
    float* out = (float*)d_out;

    const int E  = in_sizes[1] / 2;     // 800000
    const int EE = E + NNODES;          // + self loops
    const int D1 = HEADS * C1;          // 512
    const int D2 = HEADS * C2;          // 256

    // workspace carve-out (aliased where lifetimes allow; ~222 MB total)
    char* ws = (char*)d_ws;
    size_t off = 0;
    auto carve = [&](size_t bytes) -> float* {
        float* p = (float*)(ws + off);
        off += (bytes + 255) & ~(size_t)255;
        return p;
    };
    float* h1    = carve((size_t)NNODES * D1 * 4);   // reused as h2 after GEMM2
    float* agg1  = carve((size_t)NNODES * D1 * 4);   // reused as agg2
    float* ebuf  = carve((size_t)EE * HEADS * 4);
    float* asrc  = carve((size_t)NNODES * HEADS * 4);
    float* adst  = carve((size_t)NNODES * HEADS * 4);
    float* mmax  = carve((size_t)NNODES * HEADS * 4);
    float* denom = carve((size_t)NNODES * HEADS * 4);
    float* h2   = h1;
    float* agg2 = agg1;

    const int B  = 256;
    const int NH = NNODES * HEADS;
    const int EH = EE * HEADS;
    const int mTiles = NNODES / 16;                  // 3125

    // ---------------- layer 1 ----------------
    {   // h1 = x @ W1   (50000x256 @ 256x512), wave strip 16x64
        int tiles = mTiles * (D1 / 64);
        gemm16_wmma_f32<4, false><<<cdiv(tiles, 8), B, 0, stream>>>(
            x, W1, nullptr, h1, NNODES, 256, D1, D1);
    }
    attn_scores<<<cdiv((long long)NH * 32, B), B, 0, stream>>>(h1, a_src1, a_dst1,
                                                               asrc, adst, NNODES, C1);
    fill_f32<<<cdiv(NH, B), B, 0, stream>>>(mmax, -__builtin_inff(), (size_t)NH);
    fill_f32<<<cdiv(NH, B), B, 0, stream>>>(denom, 0.f, (size_t)NH);
    fill_f32<<<cdiv((long long)NNODES * D1, B), B, 0, stream>>>(agg1, 0.f,
                                                                (size_t)NNODES * D1);
    edge_score_max<<<cdiv(EH, B), B, 0, stream>>>(ei, asrc, adst, ebuf, mmax, E, EE);
    edge_exp_sum  <<<cdiv(EH, B), B, 0, stream>>>(ei, mmax, ebuf, denom, E, EE);
    edge_aggregate<<<cdiv((long long)EH * 32, B), B, 0, stream>>>(ei, h1, ebuf, denom,
                                                                  agg1, E, EE, C1);
    bias_elu<<<cdiv((long long)NNODES * D1, B), B, 0, stream>>>(agg1, b1,
                                                                (size_t)NNODES * D1, D1);

    // ---------------- layer 2 ----------------
    {   // h2 = agg1 @ W2   (50000x512 @ 512x256); h1 region is dead -> alias
        int tiles = mTiles * (D2 / 64);
        gemm16_wmma_f32<4, false><<<cdiv(tiles, 8), B, 0, stream>>>(
            agg1, W2, nullptr, h2, NNODES, D1, D2, D2);
    }
    attn_scores<<<cdiv((long long)NH * 32, B), B, 0, stream>>>(h2, a_src2, a_dst2,
                                                               asrc, adst, NNODES, C2);
    fill_f32<<<cdiv(NH, B), B, 0, stream>>>(mmax, -__builtin_inff(), (size_t)NH);
    fill_f32<<<cdiv(NH, B), B, 0, stream>>>(denom, 0.f, (size_t)NH);
    fill_f32<<<cdiv((long long)NNODES * D2, B), B, 0, stream>>>(agg2, 0.f,
                                                                (size_t)NNODES * D2);
    edge_score_max<<<cdiv(EH, B), B, 0, stream>>>(ei, asrc, adst, ebuf, mmax, E, EE);
    edge_exp_sum  <<<cdiv(EH, B), B, 0, stream>>>(ei, mmax, ebuf, denom, E, EE);
    edge_aggregate<<<cdiv((long long)EH * 32, B), B, 0, stream>>>(ei, h2, ebuf, denom,
                                                                  agg2, E, EE, C2);
    bias_elu<<<cdiv((long long)NNODES * D2, B), B, 0, stream>>>(agg2, b2,
                                                                (size_t)NNODES * D2, D2);

    // ---------------- classifier ----------------
    {   // out = agg2 @ Wf + bf   (50000x256 @ 256x50), N padded to 64, guarded
        int tiles = mTiles * 1;                      // one 16x64 strip per m-tile
        gemm16_wmma_f32<4, true><<<cdiv(tiles, 8), B, 0, stream>>>(
            agg2, Wf, bf, out, NNODES, D2, 50, 64);
    }
}